// StatDenoiser_55078660604460
// MI455X (gfx1250) — compile-verified
//
#include <hip/hip_runtime.h>
#include <cstdint>

#define RADIUS 2
#define TILE   8
#define HALO   (TILE + 2 * RADIUS)   // 12
#define TDIM   4
#define NCH    18                    // 9 guidance + 3 estimands + 3 variance + 3 noisy
#define PLANE  (128 * 128 * 4)
#define SOFF(ch, pos) ((((ch) * (HALO * HALO)) + (pos)) * TDIM)

typedef float v4f __attribute__((ext_vector_type(4)));

// Neighbor t-column shifted by dt = k-2; zero-padded lanes (matches reference padding).
// Components are individual VGPRs, so with constant k this folds to pure register renaming.
__device__ __forceinline__ v4f shiftv(const v4f g, const int k) {
    switch (k) {
        case 0:  return (v4f){0.0f, 0.0f, g.x,  g.y };   // dt = -2
        case 1:  return (v4f){0.0f, g.x,  g.y,  g.z };   // dt = -1
        case 2:  return g;                               // dt =  0
        case 3:  return (v4f){g.y,  g.z,  g.w,  0.0f};   // dt = +1
        default: return (v4f){g.z,  g.w,  0.0f, 0.0f};   // dt = +2
    }
}

__global__ __launch_bounds__(256) void statden_kernel(
    const float* __restrict__ noisy, const float* __restrict__ guid,
    const float* __restrict__ est,   const float* __restrict__ var,
    float* __restrict__ out)
{
    __shared__ __align__(16) float smem[NCH * HALO * HALO * TDIM];   // 41472 B

    constexpr float GAMMA_W = 3.3257199f;
    constexpr float G2      = GAMMA_W * GAMMA_W;               // t<=gamma <=> d^2 <= G2*(vi+vj)
    constexpr float NHL2E   = -0.5f * 1.44269504088896340736f; // fold exp(-0.5*x) -> exp2
    const float SIGE[9] = {0.1f * NHL2E, 0.1f * NHL2E, 0.1f * NHL2E,
                           50.f * NHL2E, 50.f * NHL2E, 50.f * NHL2E,
                           10.f * NHL2E, 10.f * NHL2E, 10.f * NHL2E};

    const int tid = threadIdx.x;
    const int h0  = blockIdx.y * TILE;
    const int w0  = blockIdx.x * TILE;

    // ---- Stage 18-channel halo tile into LDS via CDNA5 async global->LDS ----
    for (int i = tid; i < NCH * HALO * HALO; i += 256) {
        const int ch = i / (HALO * HALO);
        const int r  = i - ch * (HALO * HALO);
        const int hh = r / HALO;
        const int ww = r - hh * HALO;
        const int gh = h0 + hh - RADIUS;
        const int gw = w0 + ww - RADIUS;
        float* dst = &smem[(size_t)i * TDIM];
        const float* base;
        if (ch < 9)       base = guid  + (size_t)ch * PLANE;
        else if (ch < 12) base = est   + (size_t)(ch - 9)  * PLANE;
        else if (ch < 15) base = var   + (size_t)(ch - 12) * PLANE;
        else              base = noisy + (size_t)(ch - 15) * PLANE;
        if ((unsigned)gh < 128u && (unsigned)gw < 128u) {
            const float* src = base + ((((size_t)gh * 128) + gw) << 2);
            const uint32_t lds_off = (uint32_t)(uintptr_t)dst;
            asm volatile("global_load_async_to_lds_b128 %0, %1, off"
                         :: "v"(lds_off), "v"(src) : "memory");
        } else {
            *reinterpret_cast<float4*>(dst) = make_float4(0.f, 0.f, 0.f, 0.f);
        }
    }
    asm volatile("s_wait_asynccnt 0x0" ::: "memory");
    __syncthreads();

    // ---- Compute: 4 threads (chunks) per pixel, each covers 1/4 of 25 (dh,dw) ----
    const int chunk = tid & 3;
    const int pix   = tid >> 2;      // 0..63
    const int hl    = pix >> 3;      // 0..7
    const int wl    = pix & 7;       // 0..7

    v4f gi4[9], ec4[3], vc4[3];
    {
        const int cpos = (hl + RADIUS) * HALO + (wl + RADIUS);
#pragma unroll
        for (int c = 0; c < 9; ++c)
            gi4[c] = *reinterpret_cast<const v4f*>(&smem[SOFF(c, cpos)]);
#pragma unroll
        for (int c = 0; c < 3; ++c) {
            ec4[c] = *reinterpret_cast<const v4f*>(&smem[SOFF(9 + c, cpos)]);
            vc4[c] = *reinterpret_cast<const v4f*>(&smem[SOFF(12 + c, cpos)]);
        }
    }

    v4f aw4 = {0.f, 0.f, 0.f, 0.f};
    v4f a04 = {0.f, 0.f, 0.f, 0.f};
    v4f a14 = {0.f, 0.f, 0.f, 0.f};
    v4f a24 = {0.f, 0.f, 0.f, 0.f};

    for (int p = chunk; p < 25; p += 4) {
        const int dh   = p / 5;
        const int dw   = p - dh * 5;
        const int npos = (hl + dh) * HALO + (wl + dw);

        v4f gj4[9], ej4[3], vj4[3], nj4[3];
#pragma unroll
        for (int c = 0; c < 9; ++c)
            gj4[c] = *reinterpret_cast<const v4f*>(&smem[SOFF(c, npos)]);
#pragma unroll
        for (int c = 0; c < 3; ++c) {
            ej4[c] = *reinterpret_cast<const v4f*>(&smem[SOFF(9 + c, npos)]);
            vj4[c] = *reinterpret_cast<const v4f*>(&smem[SOFF(12 + c, npos)]);
            nj4[c] = *reinterpret_cast<const v4f*>(&smem[SOFF(15 + c, npos)]);
        }

#pragma unroll
        for (int k = 0; k < 5; ++k) {
            // ---- packed bilateral weight across the 4 t-lanes ----
            v4f dist = {0.f, 0.f, 0.f, 0.f};
#pragma unroll
            for (int c = 0; c < 9; ++c) {
                const v4f gjs = shiftv(gj4[c], k);
                const v4f dd  = gjs - gi4[c];
                dist += (dd * dd) * SIGE[c];     // contracts to v_pk_fma_f32
            }

            // ---- packed membership: m = d^2 - G2*(vi+vj) <= 0 ----
            bool okl[4] = {true, true, true, true};
#pragma unroll
            for (int c = 0; c < 3; ++c) {
                const v4f ejs = shiftv(ej4[c], k);
                const v4f vjs = shiftv(vj4[c], k);
                const v4f d   = ec4[c] - ejs;
                const v4f s   = vc4[c] + vjs;
                const v4f m   = d * d - s * G2;  // contracts to pk fma
#pragma unroll
                for (int t = 0; t < 4; ++t) {
                    bool pass = (m[t] <= 0.0f);
                    // zero-variance override: w forced to 1 (excluded) when ei!=ej
                    if (((vc4[c][t] == 0.0f) || (vjs[t] == 0.0f)) && (d[t] != 0.0f))
                        pass = false;
                    okl[t] = okl[t] && pass;
                }
            }

            v4f fw;
            fw.x = okl[0] ? __builtin_amdgcn_exp2f(dist.x) : 0.0f;
            fw.y = okl[1] ? __builtin_amdgcn_exp2f(dist.y) : 0.0f;
            fw.z = okl[2] ? __builtin_amdgcn_exp2f(dist.z) : 0.0f;
            fw.w = okl[3] ? __builtin_amdgcn_exp2f(dist.w) : 0.0f;

            aw4 += fw;
            a04 += fw * shiftv(nj4[0], k);       // contracts to v_pk_fma_f32
            a14 += fw * shiftv(nj4[1], k);
            a24 += fw * shiftv(nj4[2], k);
        }
    }

    // ---- Reduce the 4 chunk-lanes (adjacent lanes in a wave32) ----
    float aw[4] = {aw4.x, aw4.y, aw4.z, aw4.w};
    float a0[4] = {a04.x, a04.y, a04.z, a04.w};
    float a1[4] = {a14.x, a14.y, a14.z, a14.w};
    float a2[4] = {a24.x, a24.y, a24.z, a24.w};
#pragma unroll
    for (int t = 0; t < 4; ++t) {
        aw[t] += __shfl_xor(aw[t], 1, 32); aw[t] += __shfl_xor(aw[t], 2, 32);
        a0[t] += __shfl_xor(a0[t], 1, 32); a0[t] += __shfl_xor(a0[t], 2, 32);
        a1[t] += __shfl_xor(a1[t], 1, 32); a1[t] += __shfl_xor(a1[t], 2, 32);
        a2[t] += __shfl_xor(a2[t], 1, 32); a2[t] += __shfl_xor(a2[t], 2, 32);
    }

    if (chunk == 0) {
        const int gh = h0 + hl;
        const int gw = w0 + wl;
        const size_t pbase = (((size_t)gh * 128) + gw) * 4;
        float inv[4];
#pragma unroll
        for (int t = 0; t < 4; ++t) inv[t] = 1.0f / fmaxf(aw[t], 1e-12f);
        *reinterpret_cast<float4*>(out + 0 * PLANE + pbase) =
            make_float4(a0[0] * inv[0], a0[1] * inv[1], a0[2] * inv[2], a0[3] * inv[3]);
        *reinterpret_cast<float4*>(out + 1 * PLANE + pbase) =
            make_float4(a1[0] * inv[0], a1[1] * inv[1], a1[2] * inv[2], a1[3] * inv[3]);
        *reinterpret_cast<float4*>(out + 2 * PLANE + pbase) =
            make_float4(a2[0] * inv[0], a2[1] * inv[1], a2[2] * inv[2], a2[3] * inv[3]);
    }
}

extern "C" void kernel_launch(void* const* d_in, const int* in_sizes, int n_in,
                              void* d_out, int out_size, void* d_ws, size_t ws_size,
                              hipStream_t stream) {
    const float* noisy = (const float*)d_in[0];
    const float* guid  = (const float*)d_in[1];
    const float* est   = (const float*)d_in[2];
    const float* var   = (const float*)d_in[3];
    float* out = (float*)d_out;
    dim3 grid(128 / TILE, 128 / TILE, 1);
    hipLaunchKernelGGL(statden_kernel, grid, dim3(256), 0, stream,
                       noisy, guid, est, var, out);
}